// GCNLayer_25142738550917
// MI455X (gfx1250) — compile-verified
//
#include <hip/hip_runtime.h>
#include <hip/hip_bf16.h>
#include <math.h>

// ---------------------------------------------------------------------------
// 3-layer GCN (DGL GraphConv, norm='both', self-loops) for MI455X / gfx1250.
//   N = 100000 nodes, F = 128 features (hidden 128, out 64), E edges.
// Strategy:
//   * degrees via f32 atomics (init 1.0 covers self-loop)
//   * per layer: scale rows by d_out^-1/2 (also seeds agg with self-loop msg),
//     edge scatter-add with one wave per edge (agg buffer is L2-resident),
//     then fused (agg * d_in^-1/2) @ W + b [+ReLU] using V_WMMA_F32_16X16X4_F32
//     with the A tile staged in LDS (conflict-free padded layout) and shared
//     by all 8 column-tile waves of the block.
// ---------------------------------------------------------------------------

typedef float v2f __attribute__((ext_vector_type(2)));
typedef float v8f __attribute__((ext_vector_type(8)));

#define GCN_N 100000
#define GCN_F 128
#define LDSROW 132   // 128 + 4 pad; 132 mod 64 == 4 -> b64 reads hit 64 distinct banks

// ---- degree kernels -------------------------------------------------------

__global__ void deg_init_kernel(float* __restrict__ deg_out,
                                float* __restrict__ deg_in, int n) {
    int i = blockIdx.x * blockDim.x + threadIdx.x;
    if (i < n) { deg_out[i] = 1.0f; deg_in[i] = 1.0f; }  // self-loop counted
}

__global__ void deg_accum_kernel(const int* __restrict__ src,
                                 const int* __restrict__ dst,
                                 float* __restrict__ deg_out,
                                 float* __restrict__ deg_in, int e) {
    int i = blockIdx.x * blockDim.x + threadIdx.x;
    if (i < e) {
        atomicAdd(&deg_out[src[i]], 1.0f);
        atomicAdd(&deg_in[dst[i]], 1.0f);
    }
}

__global__ void deg_rsqrt_kernel(float* __restrict__ deg_out,
                                 float* __restrict__ deg_in, int n) {
    int i = blockIdx.x * blockDim.x + threadIdx.x;
    if (i < n) {
        deg_out[i] = rsqrtf(fmaxf(deg_out[i], 1.0f));
        deg_in[i]  = rsqrtf(fmaxf(deg_in[i], 1.0f));
    }
}

// ---- per-layer kernels ----------------------------------------------------

// hs[n][f] = in[n][f] * out_isqrt[n];  agg seeded with hs (self-loop message).
// One thread per float4 chunk: idx in [0, N*F/4), node = idx / (F/4).
__global__ void scale_init_kernel(const float* __restrict__ in,
                                  const float* __restrict__ out_isqrt,
                                  float* __restrict__ hs,
                                  float* __restrict__ agg, int nchunks) {
    int idx = blockIdx.x * blockDim.x + threadIdx.x;
    if (idx >= nchunks) return;
    int node = idx >> 5;                       // F/4 == 32 chunks per node
    float s = out_isqrt[node];
    float4 v = ((const float4*)in)[idx];
    v.x *= s; v.y *= s; v.z *= s; v.w *= s;
    ((float4*)hs)[idx]  = v;
    ((float4*)agg)[idx] = v;
}

// One wave (32 lanes) per edge; each lane moves 4 consecutive floats.
// agg (51 MB) sits in the 192 MB L2, so the f32 atomics resolve in L2.
__global__ void edge_agg_kernel(const int* __restrict__ src,
                                const int* __restrict__ dst,
                                const float* __restrict__ hs,
                                float* __restrict__ agg, int e) {
    int t = blockIdx.x * blockDim.x + threadIdx.x;
    int edge = t >> 5;
    int lane = t & 31;
    if (edge >= e) return;
    int s = src[edge];
    int d = dst[edge];
    const float4 v = *(const float4*)(hs + (size_t)s * GCN_F + lane * 4);
    float* dp = agg + (size_t)d * GCN_F + lane * 4;
    atomicAdd(dp + 0, v.x);
    atomicAdd(dp + 1, v.y);
    atomicAdd(dp + 2, v.z);
    atomicAdd(dp + 3, v.w);
}

// out[16-row stripe] = relu?( (A*in_isqrt) @ W + b ), V_WMMA_F32_16X16X4_F32.
// One block of (NC/16) waves per 16-row stripe; wave w owns column tile w.
// The scaled A tile (16x128) is staged once in LDS (coalesced float4 global
// loads) and shared by all waves; padded row stride keeps ds_load_b64
// conflict-free across all 64 banks.
// A frag (16x4 f32, 2 VGPRs): lane l: M=l&15, K = k0 + 2*(l>>4) + {0,1}.
// B frag (4x16 f32, 2 VGPRs): lane l: N=l&15, K = k0 + 2*(l>>4) + {0,1}.
// C/D (8 VGPRs): VGPR v -> M = v + 8*(l>>4), N = l&15.
__global__ void gemm_wmma_kernel(const float* __restrict__ A,   // [N, 128] agg
                                 const float* __restrict__ in_isqrt, // [N]
                                 const float* __restrict__ W,   // [128, NC] row-major
                                 const float* __restrict__ bias,// [NC]
                                 float* __restrict__ out,       // [N, NC]
                                 int NC, int do_relu) {
    const int K = GCN_F;
    __shared__ float sA[16 * LDSROW];

    const int tileM = blockIdx.x;
    const int tid   = threadIdx.x;
    const int lane  = tid & 31;
    const int wv    = tid >> 5;        // wave id == column tile
    const int half  = lane >> 4;       // 0 or 1
    const int l16   = lane & 15;

    // Cooperative, coalesced stage of the scaled A tile into LDS.
    for (int i = tid; i < 16 * (K / 4); i += blockDim.x) {
        const int row = i >> 5;        // 32 float4 chunks per row
        const int c4  = i & 31;
        const float s = in_isqrt[tileM * 16 + row];
        float4 v = ((const float4*)(A + (size_t)(tileM * 16 + row) * K))[c4];
        v.x *= s; v.y *= s; v.z *= s; v.w *= s;
        *(float4*)(&sA[row * LDSROW + c4 * 4]) = v;   // 528B row stride: 16B aligned
    }
    __syncthreads();

    const int colB = wv * 16 + l16;
    const float* __restrict__ sArow = &sA[l16 * LDSROW];

    v8f c = {};
#pragma unroll 8
    for (int kk = 0; kk < K; kk += 4) {
        const int k0 = kk + 2 * half;
        v2f a, b;
        a.x = sArow[k0];               // ds_load (bank-conflict-free)
        a.y = sArow[k0 + 1];
        b.x = W[(size_t)k0       * NC + colB];   // lane-consecutive: coalesced
        b.y = W[(size_t)(k0 + 1) * NC + colB];
        // (neg_a, A, neg_b, B, c_mod, C, reuse_a, reuse_b)
        c = __builtin_amdgcn_wmma_f32_16x16x4_f32(false, a, false, b,
                                                  (short)0, c, false, false);
    }

    const float bcol = bias[colB];
#pragma unroll
    for (int v = 0; v < 8; ++v) {
        const int m = v + 8 * half;
        float val = c[v] + bcol;
        if (do_relu) val = fmaxf(val, 0.0f);
        out[(size_t)(tileM * 16 + m) * NC + colB] = val;
    }
}

// ---------------------------------------------------------------------------

extern "C" void kernel_launch(void* const* d_in, const int* in_sizes, int n_in,
                              void* d_out, int out_size, void* d_ws, size_t ws_size,
                              hipStream_t stream) {
    const float* feat = (const float*)d_in[0];
    const int*   src  = (const int*)d_in[1];
    const int*   dst  = (const int*)d_in[2];
    const float* W1   = (const float*)d_in[3];
    const float* b1   = (const float*)d_in[4];
    const float* W2   = (const float*)d_in[5];
    const float* b2   = (const float*)d_in[6];
    const float* W3   = (const float*)d_in[7];
    const float* b3   = (const float*)d_in[8];

    const int N = GCN_N;
    const int F = GCN_F;
    const int E = in_sizes[1];

    // Workspace layout (floats): out_isqrt | in_isqrt | buf0 | buf1 | buf2
    float* p = (float*)d_ws;
    float* out_isqrt = p;                 p += N;
    float* in_isqrt  = p;                 p += N;
    float* buf0 = p;                      p += (size_t)N * F;   // h_scaled
    float* buf1 = p;                      p += (size_t)N * F;   // agg
    float* buf2 = p;                      p += (size_t)N * F;   // layer output

    const int B = 256;
    const int nchunks = N * (F / 4);      // float4 chunks per feature matrix
    const int edge_blocks = (E * 32 + B - 1) / B;

    // Degrees (self-loop folded into init value 1.0).
    deg_init_kernel<<<(N + B - 1) / B, B, 0, stream>>>(out_isqrt, in_isqrt, N);
    deg_accum_kernel<<<(E + B - 1) / B, B, 0, stream>>>(src, dst, out_isqrt, in_isqrt, E);
    deg_rsqrt_kernel<<<(N + B - 1) / B, B, 0, stream>>>(out_isqrt, in_isqrt, N);

    const int stripes = N / 16;           // 6250 16-row stripes

    // Layer 1: feat -> buf2 (relu), NC=128 -> 8 waves/block
    scale_init_kernel<<<(nchunks + B - 1) / B, B, 0, stream>>>(feat, out_isqrt, buf0, buf1, nchunks);
    edge_agg_kernel<<<edge_blocks, B, 0, stream>>>(src, dst, buf0, buf1, E);
    gemm_wmma_kernel<<<stripes, 32 * (F / 16), 0, stream>>>(buf1, in_isqrt, W1, b1, buf2, F, 1);

    // Layer 2: buf2 -> buf2 (relu)
    scale_init_kernel<<<(nchunks + B - 1) / B, B, 0, stream>>>(buf2, out_isqrt, buf0, buf1, nchunks);
    edge_agg_kernel<<<edge_blocks, B, 0, stream>>>(src, dst, buf0, buf1, E);
    gemm_wmma_kernel<<<stripes, 32 * (F / 16), 0, stream>>>(buf1, in_isqrt, W2, b2, buf2, F, 1);

    // Layer 3: buf2 -> d_out (no relu, NC=64 -> 4 waves/block)
    scale_init_kernel<<<(nchunks + B - 1) / B, B, 0, stream>>>(buf2, out_isqrt, buf0, buf1, nchunks);
    edge_agg_kernel<<<edge_blocks, B, 0, stream>>>(src, dst, buf0, buf1, E);
    gemm_wmma_kernel<<<stripes, 32 * (64 / 16), 0, stream>>>(buf1, in_isqrt, W3, b3, (float*)d_out, 64, 0);
}